// ExNASWrapper_59700045414555
// MI455X (gfx1250) — compile-verified
//
#include <hip/hip_runtime.h>
#include <hip/hip_bf16.h>

// ---------------------------------------------------------------------------
// Problem constants (mirror the reference)
// ---------------------------------------------------------------------------
#define BATCH 128
#define CIN   3
#define C1    64
#define C2    256
#define NCLS  1000
#define KCONV 48                 // _n_keep(256, 0.2, 0.05, 24)
#define KEFF  (KCONV * 256)      // 12288: K_FC=12451 > 12288 => FC == dense GEMM
                                 // over the selected channels (exactly; see notes)
#define KSTEPS (KEFF / 4)        // 3072 k-steps of 4
#define FCW_STRIDE 65536         // fc_w row stride (C2*16*16)

typedef __attribute__((ext_vector_type(2))) float v2f;
typedef __attribute__((ext_vector_type(8))) float v8f;

// ---------------------------------------------------------------------------
// Kernel 1: conv1 (3x3, pad 1) + bias + ReLU.  One thread per output element.
// ---------------------------------------------------------------------------
__global__ void conv1_relu(const float* __restrict__ x0,
                           const float* __restrict__ w1,
                           const float* __restrict__ b1,
                           float* __restrict__ x1) {
  int t  = blockIdx.x * blockDim.x + threadIdx.x;
  int hw = t & 1023;
  int co = (t >> 10) & 63;
  int b  = t >> 16;
  int h  = hw >> 5;
  int w  = hw & 31;

  const float* wp = w1 + co * 27;            // uniform within block
  const float* xb = x0 + (size_t)b * (CIN * 1024);
  float acc = b1[co];
#pragma unroll
  for (int ci = 0; ci < CIN; ++ci) {
    const float* xc = xb + ci * 1024;
    const float* wc = wp + ci * 9;
#pragma unroll
    for (int dh = -1; dh <= 1; ++dh) {
      int hh = h + dh;
#pragma unroll
      for (int dw = -1; dw <= 1; ++dw) {
        int ww2 = w + dw;
        float xv = (hh >= 0 && hh < 32 && ww2 >= 0 && ww2 < 32)
                       ? xc[hh * 32 + ww2] : 0.0f;
        acc = fmaf(xv, wc[(dh + 1) * 3 + (dw + 1)], acc);
      }
    }
  }
  x1[t] = fmaxf(acc, 0.0f);
}

// ---------------------------------------------------------------------------
// Kernel 2: sig[c] = mean_{b,h,w} |x1[b,c,h,w]| — deterministic block reduce.
// ---------------------------------------------------------------------------
__global__ void chan_absmean(const float* __restrict__ x1,
                             float* __restrict__ sig) {
  int c   = blockIdx.x;
  int tid = threadIdx.x;
  float s = 0.0f;
  for (int i = tid; i < BATCH * 1024; i += 256) {
    int b   = i >> 10;
    int off = i & 1023;
    s += fabsf(x1[(((size_t)b * 64 + c) << 10) + off]);
  }
  __shared__ float red[256];
  red[tid] = s;
  __syncthreads();
  for (int st = 128; st > 0; st >>= 1) {
    if (tid < st) red[tid] += red[tid + st];
    __syncthreads();
  }
  if (tid == 0) sig[c] = red[0] * (1.0f / (128.0f * 1024.0f));
}

// ---------------------------------------------------------------------------
// Kernel 3: scores = softplus(gate_w @ sig); exact jax.lax.top_k(48):
// descending scores, ties broken by ascending index (rank < 48 -> slot).
// ---------------------------------------------------------------------------
__global__ void gate_topk(const float* __restrict__ gate_w,
                          const float* __restrict__ sig,
                          int* __restrict__ idxc) {
  __shared__ float ssig[C1];
  __shared__ float sc[C2];
  int t = threadIdx.x;                       // 0..255, one per gate row
  if (t < C1) ssig[t] = sig[t];
  __syncthreads();
  float acc = 0.0f;
#pragma unroll 8
  for (int c = 0; c < C1; ++c) acc = fmaf(gate_w[t * C1 + c], ssig[c], acc);
  float sp = fmaxf(acc, 0.0f) + log1pf(expf(-fabsf(acc)));  // logaddexp(x,0)
  sc[t] = sp;
  __syncthreads();
  float mine = sc[t];
  int rank = 0;
  for (int j = 0; j < C2; ++j) {
    float o = sc[j];
    rank += (o > mine) || (o == mine && j < t);
  }
  if (rank < KCONV) idxc[rank] = t;
}

// ---------------------------------------------------------------------------
// Kernel 4: conv2 (48 selected filters) + bias + ReLU + 2x2 maxpool, fused.
// Output is written DIRECTLY in WMMA A-fragment order ("ypoolF"):
//   record (mt, kg) = 64 pair-slots: slot[lane*2 + e] = A[mt*16 + (lane&15)]
//                                                     [4*kg + (lane>>4)*2 + e]
// so the FC kernel's A fragment load is one coalesced global_load_b64/lane.
// A[m][col]: m = batch row, col = kc*256 + pooledPixel.
// ---------------------------------------------------------------------------
#define KPB 8
__global__ void conv2_pool(const float* __restrict__ x1,
                           const float* __restrict__ w2,
                           const float* __restrict__ b2,
                           const int* __restrict__ idxc,
                           float* __restrict__ ypoolF) {
  __shared__ __align__(16) float plane[1024];
  __shared__ float wl[KPB * 9];
  __shared__ int   chs[KPB];
  __shared__ float bs[KPB];

  int tid = threadIdx.x;
  int kc0 = blockIdx.x * KPB;                // 0,8,...,40
  int b   = blockIdx.y;

  if (tid < KPB) {
    int ch = idxc[kc0 + tid];
    chs[tid] = ch;
    bs[tid]  = b2[ch];
  }
  __syncthreads();

  int ph = tid >> 4, pw = tid & 15;
  int h0 = ph * 2, w0 = pw * 2;

  float acc[KPB][4];
#pragma unroll
  for (int k = 0; k < KPB; ++k)
#pragma unroll
    for (int s = 0; s < 4; ++s) acc[k][s] = 0.0f;

  for (int c = 0; c < C1; ++c) {
    __syncthreads();
    const float* src = x1 + (((size_t)b * 64 + c) << 10);
    *(float4*)&plane[tid * 4] = *(const float4*)&src[tid * 4];
    if (tid < KPB * 9) {
      int k = tid / 9, tap = tid % 9;
      wl[tid] = w2[(size_t)chs[k] * 576 + c * 9 + tap];
    }
    __syncthreads();
#pragma unroll
    for (int tap = 0; tap < 9; ++tap) {
      int dh = tap / 3 - 1, dw = tap % 3 - 1;
      float xv[4];
#pragma unroll
      for (int s = 0; s < 4; ++s) {
        int hh = h0 + (s >> 1) + dh;
        int w2c = w0 + (s & 1) + dw;
        xv[s] = (hh >= 0 && hh < 32 && w2c >= 0 && w2c < 32)
                    ? plane[hh * 32 + w2c] : 0.0f;
      }
#pragma unroll
      for (int k = 0; k < KPB; ++k) {
        float wv = wl[k * 9 + tap];
#pragma unroll
        for (int s = 0; s < 4; ++s) acc[k][s] = fmaf(wv, xv[s], acc[k][s]);
      }
    }
  }

  // Store in fragment-ready layout.
  int mt   = b >> 4;
  int r    = b & 15;                         // row within 16-row M tile
  int sub  = tid & 3;                        // col & 3 (tid == low 8 bits of col)
  int lane = r | ((sub & 2) << 3);           // bit4 of lane = (col&3)>>1
  int e    = sub & 1;                        // pair element
#pragma unroll
  for (int k = 0; k < KPB; ++k) {
    float bias = bs[k];
    float v0 = fmaxf(acc[k][0] + bias, 0.0f);
    float v1 = fmaxf(acc[k][1] + bias, 0.0f);
    float v2 = fmaxf(acc[k][2] + bias, 0.0f);
    float v3 = fmaxf(acc[k][3] + bias, 0.0f);
    float m  = fmaxf(fmaxf(v0, v1), fmaxf(v2, v3));
    int col = (kc0 + k) * 256 + tid;
    int kg  = col >> 2;
    ypoolF[(((size_t)mt * KSTEPS + kg) << 6) + (lane << 1) + e] = m;
  }
}

// ---------------------------------------------------------------------------
// Kernel 5: FC GEMM via V_WMMA_F32_16X16X4_F32 (exact f32 math).
//   out(128 x 1000) = A(128 x 12288) * B + fc_b,
//   B[j, n] = fc_w[n, idx_c[j>>8]*256 + (j&255)].
// Block = 256 thr = 8 waves; block owns one 16-wide N tile, wave w owns
// M-tile [16w, 16w+16).  Per selected channel the block stages the 16 KB
// fc_w slab (16 rows x 1KB contiguous) into LDS pair-packed, so each wave's
// B fragment is ONE conflict-free ds_load_b64 and the (identical across all
// 8 waves) fc_w data is fetched from HBM exactly once per block.
// A fragments are one coalesced global_load_b64 (fragment-ready layout).
// N tail: clamp row address; bad D columns are simply never stored.
// EXEC is all-ones across every WMMA.
// ---------------------------------------------------------------------------
__global__ void fc_wmma(const float* __restrict__ ypoolF,
                        const float* __restrict__ fc_w,
                        const float* __restrict__ fc_b,
                        const int* __restrict__ idxc,
                        float* __restrict__ out) {
  __shared__ int chs[KCONV];
  __shared__ __align__(16) float ldsB[4096];   // 128 pairs x 16 cols x 2
  int tid = threadIdx.x;
  if (tid < KCONV) chs[tid] = idxc[tid];
  __syncthreads();

  int wave = tid >> 5;
  int lane = tid & 31;
  int l15  = lane & 15;
  int kh   = lane >> 4;                      // 0: K pair {0,1}, 1: {2,3}
  int n0   = blockIdx.x << 4;                // 0..992

  // Staging assignment: thread -> (row j of N tile, 16-col strip).
  int jrow = tid >> 4;                       // 0..15
  int p0   = (tid & 15) << 4;                // 0,16,...,240
  int nRaw = n0 + jrow;
  int nrow = nRaw < NCLS ? nRaw : (NCLS - 1);        // clamped (tail discarded)
  const float* srcRow = fc_w + (size_t)nrow * FCW_STRIDE + p0;

  // A base: record (mt=wave, kg) at float offset ((wave*KSTEPS + kg)<<6)+lane*2
  const float* aBase = ypoolF + (((size_t)wave * KSTEPS) << 6) + (lane << 1);

  v8f c = {0.f, 0.f, 0.f, 0.f, 0.f, 0.f, 0.f, 0.f};

  for (int kc = 0; kc < KCONV; ++kc) {
    int ch = chs[kc];
    __syncthreads();                          // protect previous slab's reads
    {
      const float* s = srcRow + ch * 256;
      if (kc + 1 < KCONV)                     // global_prefetch_b8 next slab
        __builtin_prefetch(srcRow + chs[kc + 1] * 256, 0, 0);
#pragma unroll
      for (int q4 = 0; q4 < 4; ++q4) {
        float4 v = *(const float4*)(s + q4 * 4);
        int p = p0 + q4 * 4;
        // pair-packed: value(p, j) -> ldsB[((p>>1)*16 + j)*2 + (p&1)]
        ldsB[((p >> 1) * 16 + jrow) * 2 + 0]       = v.x;
        ldsB[((p >> 1) * 16 + jrow) * 2 + 1]       = v.y;
        ldsB[(((p + 2) >> 1) * 16 + jrow) * 2 + 0] = v.z;
        ldsB[(((p + 2) >> 1) * 16 + jrow) * 2 + 1] = v.w;
      }
    }
    __syncthreads();

    const float* ap = aBase + (((size_t)kc * 64) << 6);
#pragma unroll 8
    for (int s4 = 0; s4 < 64; ++s4) {         // 64 k-steps of 4 per channel
      v2f a = *(const v2f*)(ap + (s4 << 6));                  // coalesced b64
      int q = (s4 << 1) + kh;                                 // pair index
      v2f bf = *(const v2f*)&ldsB[(q * 16 + l15) * 2];        // ds b64
      c = __builtin_amdgcn_wmma_f32_16x16x4_f32(
          false, a, false, bf, (short)0, c, false, false);
    }
  }

  // Epilogue: c[r] -> row 16*wave + r + 8*kh, col n0 + l15.
  int col = n0 + l15;
  if (col < NCLS) {
    float bias = fc_b[col];
    int rowBase = (wave << 4) + (kh << 3);
#pragma unroll
    for (int r = 0; r < 8; ++r)
      out[(size_t)(rowBase + r) * NCLS + col] = c[r] + bias;
  }
}

// ---------------------------------------------------------------------------
// Launcher
// ---------------------------------------------------------------------------
extern "C" void kernel_launch(void* const* d_in, const int* in_sizes, int n_in,
                              void* d_out, int out_size, void* d_ws, size_t ws_size,
                              hipStream_t stream) {
  (void)in_sizes; (void)n_in; (void)out_size; (void)ws_size;
  const float* x0     = (const float*)d_in[0];
  const float* w1     = (const float*)d_in[1];
  const float* b1     = (const float*)d_in[2];
  const float* gate_w = (const float*)d_in[3];
  const float* w2     = (const float*)d_in[4];
  const float* b2     = (const float*)d_in[5];
  const float* fc_w   = (const float*)d_in[6];
  const float* fc_b   = (const float*)d_in[7];
  float* out = (float*)d_out;

  // workspace layout (all fully overwritten each call -> no init needed)
  float* x1     = (float*)d_ws;                     // 8388608 f
  float* sig    = x1 + (size_t)BATCH * C1 * 1024;   // 64 f
  int*   idxc   = (int*)(sig + 64);                 // 48 i (pad to 64)
  float* ypoolF = (float*)(idxc + 64);              // 1572864 f

  conv1_relu<<<(BATCH * C1 * 1024) / 256, 256, 0, stream>>>(x0, w1, b1, x1);
  chan_absmean<<<C1, 256, 0, stream>>>(x1, sig);
  gate_topk<<<1, 256, 0, stream>>>(gate_w, sig, idxc);
  conv2_pool<<<dim3(KCONV / KPB, BATCH), 256, 0, stream>>>(x1, w2, b2, idxc, ypoolF);
  fc_wmma<<<(NCLS + 15) / 16, 256, 0, stream>>>(ypoolF, fc_w, fc_b, idxc, out);
}